// FaNat_42262478193253
// MI455X (gfx1250) — compile-verified
//
#include <hip/hip_runtime.h>
#include <stdint.h>

#define LOGZERO (-1.0e10f)

constexpr int kBS      = 32;
constexpr int kXMAX    = 1024;
constexpr int kVOCAB   = 5000;
constexpr int kYMAX    = 128;
constexpr int kL       = 2 * kYMAX + 1;     // 257 interleaved blank/label states
constexpr int kThreads = 288;               // 9 wave32 waves, covers 257 states
constexpr int kNBUF    = 32;                // async gather pipeline depth (ASYNCcnt is 6b)

// ---------------- dynamic LDS layout (bytes) ----------------
constexpr int kBpOff       = 0;
constexpr int kBpBytes     = (kXMAX / 4) * kL;      // 65792: 2-bit packed backptrs
constexpr int kLppOff      = kBpOff + kBpBytes;
constexpr int kLppBytes    = kNBUF * kL * 4;        // 32896: async gather ring
constexpr int kAlphaOff    = kLppOff + kLppBytes;
constexpr int kAlphaStride = kL + 4;                // 2 leading LOGZERO pads
constexpr int kAlphaBytes  = 2 * kAlphaStride * 4;  // double-buffered alpha
constexpr int kAlignOff    = kAlphaOff + kAlphaBytes;
constexpr int kAlignBytes  = kXMAX * 2;             // int16 aligned states
constexpr int kScOff       = kAlignOff + kAlignBytes;
constexpr int kSmemBytes   = kScOff + 16;           // ~103 KB / WGP (320 KB avail)

__global__ __launch_bounds__(kThreads, 1)
void ctc_viterbi_align_kernel(const float* __restrict__ ctc_out,
                              const int*   __restrict__ src_size,
                              const int*   __restrict__ ys,
                              const int*   __restrict__ ylens,
                              float*       __restrict__ out)
{
    extern __shared__ char smem[];
    uint8_t* bp    = (uint8_t*)(smem + kBpOff);
    float*   lpp   = (float*)  (smem + kLppOff);
    float*   alpha = (float*)  (smem + kAlphaOff);
    short*   alg   = (short*)  (smem + kAlignOff);
    float*   sc    = (float*)  (smem + kScOff);

    const int b   = blockIdx.x;
    const int tid = threadIdx.x;
    const int l   = (tid < kL) ? tid : (kL - 1);   // spare lanes duplicate last state
    const int T   = src_size[b];                   // effective frame count
    const int yb  = 2 * ylens[b] + 1;              // path length

    const int* ysb = ys + b * kYMAX;
    const int tok  = (l & 1) ? ysb[l >> 1] : 0;    // blank=0 interleaved
    bool same = false;
    if (l >= 2) {
        const int tok2 = ((l - 2) & 1) ? ysb[(l - 2) >> 1] : 0;
        same = (tok == tok2);                      // blocks skip transition
    }
    const bool outside = (l >= yb);

    const float*   rowb = ctc_out + (size_t)b * kXMAX * kVOCAB;
    const uint32_t goff = (uint32_t)tok * 4u;      // per-lane byte offset into row

    // alpha init (2 leading pads are permanent LOGZERO so l-1/l-2 reads are branchless)
    float* bufA = alpha;
    float* bufB = alpha + kAlphaStride;
    if (tid < 2) { bufA[tid] = LOGZERO; bufB[tid] = LOGZERO; }
    bufA[2 + l] = (l == 0) ? 0.0f : LOGZERO;

    // ---- prologue: fill the 32-deep async gather pipeline (frames 0..31) ----
    // Single-use data: non-temporal hint keeps the 655 MB stream out of L2.
    for (int t = 0; t < kNBUF; ++t) {
        const uint32_t laddr = (uint32_t)(uintptr_t)&lpp[t * kL + l];
        const float*   srow  = rowb + (size_t)t * kVOCAB;
        asm volatile("global_load_async_to_lds_b32 %0, %1, %2 th:TH_LOAD_NT"
                     :: "v"(laddr), "v"(goff), "s"(srow) : "memory");
    }
    __syncthreads();

    // ---- forward DP: 1 barrier per frame, 1 async gather issued per frame ----
    float* cur = bufA;
    float* nxt = bufB;
    uint32_t pack = 0;
    for (int t = 0; t < kXMAX; ++t) {
        // our oldest of 32 outstanding async ops (slot t%32) has landed in LDS
        asm volatile("s_wait_asynccnt 0x1f" ::: "memory");

        const float a0 = cur[2 + l];                     // stay
        const float a1 = cur[1 + l];                     // advance 1
        const float a2 = same ? LOGZERO : cur[l];        // skip blank (if labels differ)
        float m = a0; int idx = 0;
        if (a1 > m) { m = a1; idx = 1; }                 // first-max tie-break = argmax
        if (a2 > m) { m = a2; idx = 2; }

        const float lp = (t < T) ? lpp[(t & (kNBUF - 1)) * kL + l] : LOGZERO;
        const float an = (outside ? LOGZERO : m) + lp;
        nxt[2 + l] = an;

        pack |= (uint32_t)idx << ((t & 3) * 2);          // 2-bit backptr packing
        if ((t & 3) == 3) { bp[(t >> 2) * kL + l] = (uint8_t)pack; pack = 0; }

        if (t == T - 1) {                                // capture alpha_end terminals
            if (l == yb - 1) sc[0] = an;
            if (l == yb - 2) sc[1] = an;
        }

        // refill the just-consumed slot with frame t+32 (tail clamps re-gather the
        // last row into its own slot: same data, benign; keeps 1 issue/iter so the
        // wait-count stays aligned). Each LDS word has a single writer lane == its
        // reader lane, so the ring needs no barrier of its own.
        const int      tf    = (t + kNBUF < kXMAX) ? (t + kNBUF) : (kXMAX - 1);
        const uint32_t laddr = (uint32_t)(uintptr_t)&lpp[(tf & (kNBUF - 1)) * kL + l];
        const float*   srow  = rowb + (size_t)tf * kVOCAB;
        asm volatile("global_load_async_to_lds_b32 %0, %1, %2 th:TH_LOAD_NT"
                     :: "v"(laddr), "v"(goff), "s"(srow) : "memory");

        __syncthreads();                                 // publish nxt alpha
        float* tmp = cur; cur = nxt; nxt = tmp;
    }
    asm volatile("s_wait_asynccnt 0x0" ::: "memory");
    __syncthreads();

    // ---- backtrace: sequential pointer chase entirely in LDS (2-bit unpack) ----
    if (tid == 0) {
        const float s1 = sc[0];
        const float s2 = sc[1];
        out[kBS * kXMAX + b] = fmaxf(s1, s2);            // score
        const int init_state = (s1 > s2) ? (yb - 1) : (yb - 2);
        int s = 0;
        for (int t = kXMAX - 1; t >= 0; --t) {
            if (t >= T)          s = 0;
            else if (t == T - 1) s = init_state;
            else {
                const int tp1 = t + 1;
                const uint32_t byte = bp[(tp1 >> 2) * kL + s];
                s -= (int)((byte >> ((tp1 & 3) * 2)) & 3u);
            }
            alg[t] = (short)s;
        }
    }
    __syncthreads();

    // ---- tokens + collapse repeats + shift-by-one, parallel over frames ----
    // d_out layout: [32*1024] aligned tokens (stored as float values per the
    // harness "non-bf16 output -> float*" convention), then [32] scores.
    float* outrow = out + (size_t)b * kXMAX;
    for (int t = tid; t < kXMAX; t += kThreads) {
        const int st = alg[t];
        const int tk = (st & 1) ? ysb[st >> 1] : 0;
        int pv = 0;
        if (t > 0) { const int sp = alg[t - 1]; pv = (sp & 1) ? ysb[sp >> 1] : 0; }
        const int coll = (tk == pv) ? 0 : tk;            // collapse repeats
        if (t + 1 < kXMAX) outrow[t + 1] = (float)coll;  // shifted output
        if (t == 0)        outrow[0]     = 0.0f;
    }
}

extern "C" void kernel_launch(void* const* d_in, const int* in_sizes, int n_in,
                              void* d_out, int out_size, void* d_ws, size_t ws_size,
                              hipStream_t stream) {
    (void)in_sizes; (void)n_in; (void)out_size; (void)d_ws; (void)ws_size;
    const float* ctc_out  = (const float*)d_in[0];
    // d_in[1] = src_mask: redundant (mask == t < src_size), not read
    const int*   src_size = (const int*)d_in[2];
    const int*   ys       = (const int*)d_in[3];
    const int*   ylens    = (const int*)d_in[4];
    float*       out      = (float*)d_out;

    // ~103 KB dynamic LDS per block (CDNA5 WGP has 320 KB); idempotent, capture-safe.
    hipFuncSetAttribute(reinterpret_cast<const void*>(ctc_viterbi_align_kernel),
                        hipFuncAttributeMaxDynamicSharedMemorySize, kSmemBytes);

    ctc_viterbi_align_kernel<<<kBS, kThreads, kSmemBytes, stream>>>(
        ctc_out, src_size, ys, ylens, out);
}